// APPNP_Net_23390391894788
// MI455X (gfx1250) — compile-verified
//
#include <hip/hip_runtime.h>

// APPNP GNN inference for MI455X (gfx1250, wave32).
// MLP front-end uses V_WMMA_F32_16X16X4_F32 (full fp32 — the GEMM is
// HBM-bandwidth-bound, so low-precision WMMA gains nothing).
// Propagation (K=10) is L2-resident atomic scatter-add: h is 25.6MB << 192MB L2.

typedef float v2f __attribute__((ext_vector_type(2)));
typedef float v8f __attribute__((ext_vector_type(8)));

#define M_IN   256   // input feature dim
#define H_DIM  64    // hidden == output feature dim
#define LDT    66    // LDS leading dim (floats) -> bank-conflict-free layer-2 reads

// ---------------------------------------------------------------------------
// Fused 2-layer MLP: h0 = relu(x @ W1^T + b1) @ W2^T + b2
// Block: 256 threads = 8 waves. Block covers 64 rows.
// Wave w: row tile rt = w>>1 (16 rows), col tiles {32*(w&1), 32*(w&1)+16}.
// ---------------------------------------------------------------------------
__global__ __launch_bounds__(256)
void appnp_mlp_wmma(const float* __restrict__ x,
                    const float* __restrict__ W1, const float* __restrict__ b1,
                    const float* __restrict__ W2, const float* __restrict__ b2,
                    float* __restrict__ h0, int N) {
  __shared__ float t_lds[64 * LDT];

  const int lane = threadIdx.x & 31;
  const int w    = threadIdx.x >> 5;
  const int lid  = lane & 15;     // M (or N) index within tile
  const int half = lane >> 4;     // selects K pair {0,1} vs {2,3}
  const int rt   = w >> 1;        // row tile 0..3
  const int cp   = w & 1;         // col pair 0..1
  const int c0   = cp * 32;       // first col-tile base
  const int c1   = c0 + 16;       // second col-tile base

  const int row_base = blockIdx.x * 64 + rt * 16;
  int arow = row_base + lid;
  if (arow >= N) arow = N - 1;    // clamp; garbage rows masked at store

  // ---- Layer 1: [16 x 256] @ [256 x 32] via 16x16x4 f32 WMMA ----
  const float* xrow = x  + (size_t)arow * M_IN + 2 * half;
  const float* w1a  = W1 + (size_t)(c0 + lid) * M_IN + 2 * half;
  const float* w1b  = W1 + (size_t)(c1 + lid) * M_IN + 2 * half;

  v8f acc0 = {};
  v8f acc1 = {};
#pragma unroll 8
  for (int k = 0; k < M_IN; k += 4) {
    v2f a  = *(const v2f*)(xrow + k);   // A(m=lid, K=k+2*half..+1)
    v2f bA = *(const v2f*)(w1a + k);    // B(K, n=lid) = W1[n][K]
    v2f bB = *(const v2f*)(w1b + k);
    acc0 = __builtin_amdgcn_wmma_f32_16x16x4_f32(false, a, false, bA,
                                                 (short)0, acc0, false, false);
    acc1 = __builtin_amdgcn_wmma_f32_16x16x4_f32(false, a, false, bB,
                                                 (short)0, acc1, false, false);
  }

  const float bias0 = b1[c0 + lid];
  const float bias1 = b1[c1 + lid];
#pragma unroll
  for (int v = 0; v < 8; ++v) {
    const int r  = rt * 16 + v + 8 * half;   // C/D layout: VGPR v -> M = v + 8*half
    float t0 = acc0[v] + bias0; t0 = t0 > 0.f ? t0 : 0.f;
    float t1 = acc1[v] + bias1; t1 = t1 > 0.f ? t1 : 0.f;
    t_lds[r * LDT + c0 + lid] = t0;
    t_lds[r * LDT + c1 + lid] = t1;
  }
  __syncthreads();

  // ---- Layer 2: [16 x 64] @ [64 x 32] from LDS ----
  const float* ta  = t_lds + (rt * 16 + lid) * LDT + 2 * half;
  const float* w2a = W2 + (size_t)(c0 + lid) * H_DIM + 2 * half;
  const float* w2b = W2 + (size_t)(c1 + lid) * H_DIM + 2 * half;

  v8f d0 = {};
  v8f d1 = {};
#pragma unroll
  for (int k = 0; k < H_DIM; k += 4) {
    v2f a  = *(const v2f*)(ta + k);
    v2f bA = *(const v2f*)(w2a + k);
    v2f bB = *(const v2f*)(w2b + k);
    d0 = __builtin_amdgcn_wmma_f32_16x16x4_f32(false, a, false, bA,
                                               (short)0, d0, false, false);
    d1 = __builtin_amdgcn_wmma_f32_16x16x4_f32(false, a, false, bB,
                                               (short)0, d1, false, false);
  }

  const float ob0 = b2[c0 + lid];
  const float ob1 = b2[c1 + lid];
#pragma unroll
  for (int v = 0; v < 8; ++v) {
    const int grow = row_base + v + 8 * half;
    if (grow < N) {
      h0[(size_t)grow * H_DIM + c0 + lid] = d0[v] + ob0;
      h0[(size_t)grow * H_DIM + c1 + lid] = d1[v] + ob1;
    }
  }
}

// ---------------------------------------------------------------------------
// Degree / norm kernels
// ---------------------------------------------------------------------------
__global__ void deg_init(float* __restrict__ deg, int N) {
  int i = blockIdx.x * blockDim.x + threadIdx.x;
  if (i < N) deg[i] = 1.0f;   // self-loop
}

__global__ void deg_edges(const int* __restrict__ dst, float* __restrict__ deg, int E) {
  int e = blockIdx.x * blockDim.x + threadIdx.x;
  if (e < E) unsafeAtomicAdd(&deg[dst[e]], 1.0f);  // hardware global_atomic_add_f32
}

__global__ void deg_rsqrt(float* __restrict__ deg, int N) {
  int i = blockIdx.x * blockDim.x + threadIdx.x;
  if (i < N) deg[i] = rsqrtf(deg[i]);   // deg >= 1 always (self-loop)
}

// ---------------------------------------------------------------------------
// Propagation round kernels
// ---------------------------------------------------------------------------
// agg[i] = dinv[i]^2 * h[i]   (the self-loop message)
__global__ void prop_init(const float* __restrict__ h, const float* __restrict__ dinv,
                          float* __restrict__ agg, int total) {
  int t = blockIdx.x * blockDim.x + threadIdx.x;
  if (t < total) {
    float d = dinv[t >> 6];
    agg[t] = d * d * h[t];
  }
}

// 16 lanes per edge, 4 features each: coalesced 256B gather of h[src],
// 4x global_atomic_add_f32 into agg[dst]. Everything L2-resident.
__global__ __launch_bounds__(256)
void prop_scatter(const int* __restrict__ src, const int* __restrict__ dst,
                  const float* __restrict__ dinv, const float* __restrict__ h,
                  float* __restrict__ agg, int E) {
  int t = blockIdx.x * blockDim.x + threadIdx.x;
  int e = t >> 4;
  if (e >= E) return;
  int part = t & 15;
  int s = src[e];
  int d = dst[e];
  float nrm = dinv[s] * dinv[d];
  const float4 hv = *(const float4*)(h + (size_t)s * H_DIM + part * 4);
  float* out = agg + (size_t)d * H_DIM + part * 4;
  unsafeAtomicAdd(out + 0, nrm * hv.x);
  unsafeAtomicAdd(out + 1, nrm * hv.y);
  unsafeAtomicAdd(out + 2, nrm * hv.z);
  unsafeAtomicAdd(out + 3, nrm * hv.w);
}

// h' = (1-alpha)*agg + alpha*h0   (in-place on agg is safe; elementwise)
__global__ void prop_combine(const float* __restrict__ agg, const float* __restrict__ h0,
                             float* __restrict__ out, int total) {
  int t = blockIdx.x * blockDim.x + threadIdx.x;
  if (t < total) out[t] = 0.9f * agg[t] + 0.1f * h0[t];
}

// ---------------------------------------------------------------------------
extern "C" void kernel_launch(void* const* d_in, const int* in_sizes, int n_in,
                              void* d_out, int out_size, void* d_ws, size_t ws_size,
                              hipStream_t stream) {
  const float* x  = (const float*)d_in[0];
  const float* W1 = (const float*)d_in[1];
  const float* b1 = (const float*)d_in[2];
  const float* W2 = (const float*)d_in[3];
  const float* b2 = (const float*)d_in[4];
  const int*   ei = (const int*)d_in[5];   // harness contract: integer -> const int*

  const int N = in_sizes[0] / M_IN;        // 100000
  const int E = in_sizes[5] / 2;           // 1600000
  const int* src = ei;        // edge_index[0]
  const int* dst = ei + E;    // edge_index[1]

  const size_t feat_bytes = (size_t)N * H_DIM * sizeof(float);
  char* ws = (char*)d_ws;
  float* h0  = (float*)(ws);
  float* hA  = (float*)(ws + feat_bytes);
  float* hB  = (float*)(ws + 2 * feat_bytes);
  float* deg = (float*)(ws + 3 * feat_bytes);   // becomes dinv after rsqrt

  // 1) MLP -> h0
  appnp_mlp_wmma<<<(N + 63) / 64, 256, 0, stream>>>(x, W1, b1, W2, b2, h0, N);

  // 2) GCN symmetric norm
  deg_init <<<(N + 255) / 256, 256, 0, stream>>>(deg, N);
  deg_edges<<<(E + 255) / 256, 256, 0, stream>>>(dst, deg, E);
  deg_rsqrt<<<(N + 255) / 256, 256, 0, stream>>>(deg, N);

  // 3) K = 10 propagation rounds
  const int total = N * H_DIM;
  const int gridT = (total + 255) / 256;
  const long long sthreads = (long long)E * 16;
  const int gridS = (int)((sthreads + 255) / 256);

  float* cur = h0;
  float* bufs[2] = {hA, hB};
  for (int it = 0; it < 10; ++it) {
    float* agg = bufs[it & 1];
    prop_init   <<<gridT, 256, 0, stream>>>(cur, deg, agg, total);
    prop_scatter<<<gridS, 256, 0, stream>>>(src, dst, deg, cur, agg, E);
    float* out = (it == 9) ? (float*)d_out : agg;
    prop_combine<<<gridT, 256, 0, stream>>>(agg, h0, out, total);
    cur = out;
  }
}